// PhaseGrouping_19439021981973
// MI455X (gfx1250) — compile-verified
//
#include <hip/hip_runtime.h>
#include <math.h>

#define MAXG 16
#define COS_TH 0.8775825618903728f   // cos(0.5) : d < 0.5  <=>  dot > cos(0.5)

typedef __attribute__((ext_vector_type(2))) float v2f;
typedef __attribute__((ext_vector_type(8))) float v8f;

// ---------------------------------------------------------------------------
// Kernel 1: per (b,t) row: acc = <features[b,t,:], w_phase>; store sin/cos.
// One wave32 per row; w_phase staged in LDS once per block.
// ---------------------------------------------------------------------------
__global__ void __launch_bounds__(256)
k_phase(const float* __restrict__ feat, const float* __restrict__ w,
        float* __restrict__ sins, float* __restrict__ coss, int nrows, int D)
{
    __shared__ __align__(16) float sw[1024];
    for (int i = threadIdx.x; i < D; i += blockDim.x) sw[i] = w[i];
    __syncthreads();

    int row  = blockIdx.x * (blockDim.x >> 5) + (threadIdx.x >> 5);
    if (row >= nrows) return;
    int lane = threadIdx.x & 31;

    const float4* f4 = (const float4*)(feat + (size_t)row * D);
    const float4* w4 = (const float4*)sw;
    float acc = 0.f;
#pragma unroll
    for (int i = 0; i < 8; ++i) {            // 8 * 32 lanes * 4 = 1024
        float4 a = f4[i * 32 + lane];
        float4 b = w4[i * 32 + lane];
        acc += a.x * b.x + a.y * b.y + a.z * b.z + a.w * b.w;
    }
#pragma unroll
    for (int off = 16; off >= 1; off >>= 1)
        acc += __shfl_xor(acc, off, 32);

    if (lane == 0) {                          // sin/cos are 2pi-periodic: no mod needed
        sins[row] = sinf(acc);
        coss[row] = cosf(acc);
    }
}

// ---------------------------------------------------------------------------
// Kernel 2: sequential scan, one wave32 per sequence; lane g owns group g.
// Centroid kept as unit vector (S,C); match test and nearest-group search are
// pure dot products => zero transcendentals in the serial dependency chain.
// ---------------------------------------------------------------------------
__global__ void __launch_bounds__(32)
k_scan(const float* __restrict__ sins, const float* __restrict__ coss,
       int* __restrict__ gids_i, float* __restrict__ gids_f,
       float* __restrict__ cnt, float* __restrict__ out_num,
       float* __restrict__ out_coh, int T)
{
    int b = blockIdx.x;
    int lane = threadIdx.x;                   // 0..31 ; lanes 0..15 = groups
    float S = 0.f, C = 1.f, count = 0.f;      // centroid unit vector + scan count
    float members = 0.f, ssum = 0.f, csum = 0.f;
    int num = 0;
    const float* sp = sins + b * T;
    const float* cp = coss + b * T;

    for (int t = 0; t < T; ++t) {
        float sphi = sp[t];
        float cphi = cp[t];
        float dot  = cphi * C + sphi * S;     // cos(angular distance)
        bool active = lane < num;
        bool valid  = active && (dot > COS_TH);
        unsigned bal = (unsigned)__ballot(valid);
        bool any_valid = (bal != 0);
        int g_match = __ffs(bal) - 1;

        // argmax dot among active lanes, first-index tiebreak == argmin distance
        float key = active ? dot : -3.4e38f;
        int   ki  = lane;
#pragma unroll
        for (int off = 8; off >= 1; off >>= 1) {
            float ok = __shfl_xor(key, off, 32);
            int   oi = __shfl_xor(ki,  off, 32);
            if (ok > key || (ok == key && oi < ki)) { key = ok; ki = oi; }
        }
        int g_near = __shfl(ki, 0, 32);
        int g = any_valid ? g_match : ((num < MAXG) ? num : g_near);
        g = __shfl(g, 0, 32);                 // make uniform

        if (lane == g) {
            if (any_valid) {                  // running circular-mean update
                float y = count * S + sphi;
                float x = count * C + cphi;
                float rn = rsqrtf(x * x + y * y);
                S = y * rn; C = x * rn; count += 1.f;
            } else if (num < MAXG) {          // new group
                S = sphi; C = cphi; count = 1.f;
            }                                 // forced-nearest: centroid untouched
            members += 1.f; ssum += sphi; csum += cphi;
        }
        if (!any_valid && num < MAXG) num++;

        if (lane == 0) {
            gids_i[b * T + t] = g;
            gids_f[b * T + t] = (float)g;
        }
    }

    if (lane < MAXG) cnt[b * MAXG + lane] = members;
    float r = 0.f;
    if (lane < MAXG && members > 1.f)
        r = sqrtf(ssum * ssum + csum * csum) / members;
#pragma unroll
    for (int off = 8; off >= 1; off >>= 1) r += __shfl_xor(r, off, 32);
    if (lane == 0) {
        out_num[b] = (float)num;
        out_coh[b] = (num > 0) ? (__shfl(r, 0, 32) / (float)num) : 0.f;
    }
}

// ---------------------------------------------------------------------------
// Kernel 3: gsum[b,g,d] = sum_t onehot[b,t,g] * features[b,t,d] via
// V_WMMA_F32_16X16X4_F32.  One wave per (b, 16-wide d tile).  Feature tiles
// (64 t x 16 d) double-buffered into LDS with global_load_async_to_lds_b128.
// ---------------------------------------------------------------------------
__global__ void __launch_bounds__(128)
k_group(const float* __restrict__ feat, const int* __restrict__ gids,
        const float* __restrict__ cnt, float* __restrict__ out_gf,
        int T, int D)
{
    const int DT = D / 16;                              // 64 d-tiles
    __shared__ __align__(16) float sfeat[4][2][64 * 16]; // [wave][buf][t][n]
    __shared__ int sgid[4][64];

    int wib  = threadIdx.x >> 5;
    int lane = threadIdx.x & 31;
    int gw   = blockIdx.x * 4 + wib;
    int b     = gw / DT;
    int dtile = gw % DT;
    int d0    = dtile * 16;

    const float* fb = feat + (size_t)b * T * D + d0;    // [t][16] slice base
    const int*   gb = gids + b * T;

    unsigned lds0 = (unsigned)(size_t)(void*)&sfeat[wib][0][0];
    unsigned lds1 = (unsigned)(size_t)(void*)&sfeat[wib][1][0];

    auto issue = [&](int t0, int buf) {
        unsigned base = buf ? lds1 : lds0;
#pragma unroll
        for (int i = 0; i < 8; ++i) {                   // 64 rows x 64B = 8 x (32 lanes x 16B)
            int task = i * 32 + lane;
            int r  = task >> 2;
            int cc = task & 3;
            const float* gp = fb + (size_t)(t0 + r) * D + cc * 4;
            unsigned lp = base + (unsigned)(r * 64 + cc * 16);
            asm volatile("global_load_async_to_lds_b128 %0, %1, off"
                         :: "v"(lp), "v"(gp) : "memory");
        }
    };

    v8f acc = {};
    int half = lane >> 4;                               // K-half selector
    int m    = lane & 15;                               // group row / d column
    const int nch = T / 64;                             // 32 chunks

    issue(0, 0);
    for (int c = 0; c < nch; ++c) {
        int t0 = c * 64;
        sgid[wib][lane]      = gb[t0 + lane];           // stage 64 gids in LDS
        sgid[wib][lane + 32] = gb[t0 + lane + 32];
        if (c + 1 < nch) {
            issue(t0 + 64, (c + 1) & 1);
            asm volatile("s_wait_asynccnt 8" ::: "memory");   // chunk c complete
        } else {
            asm volatile("s_wait_asynccnt 0" ::: "memory");
        }
        const float* lf = &sfeat[wib][c & 1][0];
        const int*   lg = &sgid[wib][0];
#pragma unroll
        for (int j = 0; j < 16; ++j) {                  // 16 k-steps of K=4
            int k0 = 4 * j + 2 * half;
            int g0 = lg[k0], g1 = lg[k0 + 1];
            v2f a, bb;
            a.x  = (g0 == m) ? 1.f : 0.f;               // A: one-hot (16 x 4)
            a.y  = (g1 == m) ? 1.f : 0.f;
            bb.x = lf[k0 * 16 + m];                     // B: features (4 x 16)
            bb.y = lf[(k0 + 1) * 16 + m];
            acc = __builtin_amdgcn_wmma_f32_16x16x4_f32(
                false, a, false, bb, (short)0, acc, false, false);
        }
    }

    // C layout: VGPR r, lanes 0-15 -> M=r ; lanes 16-31 -> M=r+8 ; N = m
#pragma unroll
    for (int r = 0; r < 8; ++r) {
        int M = r + 8 * half;
        float cd = fmaxf(cnt[b * MAXG + M], 1.f);
        out_gf[((size_t)(b * MAXG + M)) * D + d0 + m] = acc[r] / cd;
    }
}

// ---------------------------------------------------------------------------
extern "C" void kernel_launch(void* const* d_in, const int* in_sizes, int n_in,
                              void* d_out, int out_size, void* d_ws, size_t ws_size,
                              hipStream_t stream)
{
    const float* feat = (const float*)d_in[0];
    const float* w    = (const float*)d_in[1];
    const int B = 16, T = 2048, D = 1024;

    float* ws   = (float*)d_ws;
    float* sins = ws;
    float* coss = ws + (size_t)B * T;
    int*   gids = (int*)(ws + (size_t)2 * B * T);
    float* cnt  = ws + (size_t)3 * B * T;

    float* out      = (float*)d_out;
    float* out_gid  = out;                         // B*T
    float* out_gf   = out + (size_t)B * T;         // B*G*D
    float* out_num  = out_gf + (size_t)B * MAXG * D;
    float* out_coh  = out_num + B;

    k_phase<<<(B * T) / 8, 256, 0, stream>>>(feat, w, sins, coss, B * T, D);
    k_scan <<<B, 32, 0, stream>>>(sins, coss, gids, out_gid, cnt, out_num, out_coh, T);
    k_group<<<(B * (D / 16)) / 4, 128, 0, stream>>>(feat, gids, cnt, out_gf, T, D);
}